// USA_25091198943307
// MI455X (gfx1250) — compile-verified
//
#include <hip/hip_runtime.h>
#include <math.h>

typedef __attribute__((ext_vector_type(2))) float v2f;
typedef __attribute__((ext_vector_type(8))) float v8f;

#define NA   16      // heads
#define SEQ  2048    // SK == SQ
#define DIM  128     // D == IDM
#define RP   4       // R padded 3 -> 4
#define TINV 0.1f    // 1 / T_ANNEAL
#define RINV (1.0f/3.0f)

// ---------------------------------------------------------------------------
// Fold the two projection matmuls: Weff[a,d,r] = sum_e Wx[e,d] * W[a,e,r]
// beff[a,r] = sum_e bx[e] * W[a,e,r].  r padded to 4 with zeros.
// ---------------------------------------------------------------------------
__global__ void weff_kernel(const float* __restrict__ Wx,   // [IDM, D] = [128,128]
                            const float* __restrict__ bx,   // [IDM]
                            const float* __restrict__ W,    // [A, IDM, 3]
                            float* __restrict__ Weff,       // [A, D, 4]
                            float* __restrict__ beff) {     // [A, 4]
    int idx = blockIdx.x * blockDim.x + threadIdx.x;        // a*128 + d
    if (idx >= NA * DIM) return;
    int a = idx >> 7;
    int d = idx & 127;
    const float* Wa = W + (size_t)a * DIM * 3;
    float s0 = 0.f, s1 = 0.f, s2 = 0.f;
    #pragma unroll 4
    for (int e = 0; e < DIM; ++e) {
        float wk = Wx[(size_t)e * DIM + d];
        s0 += wk * Wa[e * 3 + 0];
        s1 += wk * Wa[e * 3 + 1];
        s2 += wk * Wa[e * 3 + 2];
    }
    float4 o = {s0, s1, s2, 0.f};
    *(float4*)(Weff + (size_t)idx * RP) = o;
    if (d == 0) {
        float b0 = 0.f, b1 = 0.f, b2 = 0.f;
        #pragma unroll 4
        for (int e = 0; e < DIM; ++e) {
            float bv = bx[e];
            b0 += bv * Wa[e * 3 + 0];
            b1 += bv * Wa[e * 3 + 1];
            b2 += bv * Wa[e * 3 + 2];
        }
        float4 ob = {b0, b1, b2, 0.f};
        *(float4*)(beff + (size_t)a * RP) = ob;
    }
}

// ---------------------------------------------------------------------------
// Codes: one wave per (a,s) row.  C[a,s,r] = tanh((X[a,s,:]·Weff[a,:,r]+beff)/T)
// Lane l handles d = 4l..4l+3 (coalesced float4 load), shfl_xor reduction.
// ---------------------------------------------------------------------------
__global__ void codes_kernel(const float* __restrict__ X,     // [A, SEQ, D]
                             const float* __restrict__ Weff,  // [A, D, 4]
                             const float* __restrict__ beff,  // [A, 4]
                             float* __restrict__ C) {         // [A, SEQ, 4]
    const int lane = threadIdx.x & 31;
    const int wave = threadIdx.x >> 5;
    const int row  = blockIdx.x * (blockDim.x >> 5) + wave;   // 0 .. NA*SEQ-1
    const int a    = row >> 11;                               // row / 2048

    float4 xv = *(const float4*)(X + (size_t)row * DIM + lane * 4);
    const float* wp = Weff + ((size_t)a * DIM + lane * 4) * RP;

    float s0 = 0.f, s1 = 0.f, s2 = 0.f;
    const float xs[4] = {xv.x, xv.y, xv.z, xv.w};
    #pragma unroll
    for (int j = 0; j < 4; ++j) {
        float x = xs[j];
        s0 += x * wp[j * RP + 0];
        s1 += x * wp[j * RP + 1];
        s2 += x * wp[j * RP + 2];
    }
    #pragma unroll
    for (int off = 16; off > 0; off >>= 1) {
        s0 += __shfl_xor(s0, off, 32);
        s1 += __shfl_xor(s1, off, 32);
        s2 += __shfl_xor(s2, off, 32);
    }
    if (lane == 0) {
        const float* bp = beff + (size_t)a * RP;
        float4 o;
        o.x = tanhf((s0 + bp[0]) * TINV);
        o.y = tanhf((s1 + bp[1]) * TINV);
        o.z = tanhf((s2 + bp[2]) * TINV);
        o.w = 0.f;                          // zero pad lane -> K=4 dot == K=3 dot
        *(float4*)(C + (size_t)row * RP) = o;
    }
}

// ---------------------------------------------------------------------------
// Main agreement kernel: one wave computes one 16x16 output tile with a single
// V_WMMA_F32_16X16X4_F32 (K padded 3->4).  Epilogue: sigmoid(pc/T)*pc with a
// fast v_rcp_f32 (no IEEE div refinement), streamed out with non-temporal
// stores so the 268MB output doesn't sweep L2 (codes stay resident).
//
// A (16x4 f32) lane layout (ISA 7.12.2): lanes 0-15: M=lane, VGPR0=K0 VGPR1=K1;
//                                        lanes 16-31: M=lane-16, VGPR0=K2 VGPR1=K3.
// B (4x16) mirrors it with N=lane&15.  Since codes rows are [s][4] contiguous,
// both operands load a v2f at (row, hi*2) -- identical addressing.
// C/D: VGPR i -> M = i + 8*(lane>=16), N = lane&15.
// ---------------------------------------------------------------------------
__global__ void wmma_tile_kernel(const float* __restrict__ Qc,  // [A, SEQ, 4]
                                 const float* __restrict__ Kc,  // [A, SEQ, 4]
                                 float* __restrict__ out) {     // [A, SEQ, SEQ]
    const int lane = threadIdx.x & 31;
    const int wave = threadIdx.x >> 5;
    const int tile = blockIdx.x * 8 + wave;     // 8 waves / block, exact grid

    const int TPR = SEQ / 16;                   // 128 tiles per row
    const int TPH = TPR * TPR;                  // 16384 tiles per head
    const int a   = tile / TPH;
    const int t   = tile - a * TPH;
    const int tq  = t / TPR;
    const int ts  = t - tq * TPR;
    const int qbase = tq * 16;
    const int sbase = ts * 16;

    const int m  = lane & 15;
    const int hi = lane >> 4;                   // 0 -> K0/K1, 1 -> K2/K3

    v2f av = *(const v2f*)(Qc + ((size_t)a * SEQ + qbase + m) * RP + hi * 2);
    v2f bv = *(const v2f*)(Kc + ((size_t)a * SEQ + sbase + m) * RP + hi * 2);

    v8f c = {};
    c = __builtin_amdgcn_wmma_f32_16x16x4_f32(
        /*neg_a=*/false, av, /*neg_b=*/false, bv,
        /*c_mod=*/(short)0, c, /*reuse_a=*/false, /*reuse_b=*/false);

    // epilogue + store: row M = i + 8*hi, col N = m
    float* orow = out + ((size_t)a * SEQ + qbase + hi * 8) * SEQ + sbase + m;
    #pragma unroll
    for (int i = 0; i < 8; ++i) {
        float pc  = c[i] * RINV;
        // sigmoid(pc/10)*pc with fast hardware reciprocal (v_rcp_f32, ~1ulp)
        float sig = __builtin_amdgcn_rcpf(1.0f + __expf(-pc * TINV));
        __builtin_nontemporal_store(sig * pc, orow + (size_t)i * SEQ);
    }
}

// ---------------------------------------------------------------------------
extern "C" void kernel_launch(void* const* d_in, const int* in_sizes, int n_in,
                              void* d_out, int out_size, void* d_ws, size_t ws_size,
                              hipStream_t stream) {
    const float* K  = (const float*)d_in[0];   // [1,16,2048,128]
    const float* Q  = (const float*)d_in[1];   // [1,16,2048,128]
    const float* Wk = (const float*)d_in[2];   // [128,128]
    const float* bk = (const float*)d_in[3];   // [128]
    const float* Wq = (const float*)d_in[4];   // [128,128]
    const float* bq = (const float*)d_in[5];   // [128]
    const float* W  = (const float*)d_in[6];   // [16,128,3]
    float* out = (float*)d_out;                // [16,2048,2048]

    float* ws    = (float*)d_ws;
    float* WeffK = ws;                 // 16*128*4 = 8192
    float* beffK = ws + 8192;          // 64
    float* WeffQ = ws + 8256;          // 8192
    float* beffQ = ws + 16448;         // 64
    float* Kc    = ws + 16512;         // 16*2048*4 = 131072
    float* Qc    = ws + 147584;        // 131072   (total ~1.06 MB)

    weff_kernel<<<(NA * DIM + 255) / 256, 256, 0, stream>>>(Wk, bk, W, WeffK, beffK);
    weff_kernel<<<(NA * DIM + 255) / 256, 256, 0, stream>>>(Wq, bq, W, WeffQ, beffQ);

    // NA*SEQ = 32768 rows per side, 8 waves/block -> 4096 blocks
    codes_kernel<<<(NA * SEQ) / 8, 256, 0, stream>>>(K, WeffK, beffK, Kc);
    codes_kernel<<<(NA * SEQ) / 8, 256, 0, stream>>>(Q, WeffQ, beffQ, Qc);

    // 16 heads * 128*128 tiles = 262144 tiles, 8 waves/block -> 32768 blocks
    wmma_tile_kernel<<<NA * (SEQ / 16) * (SEQ / 16) / 8, 256, 0, stream>>>(Qc, Kc, out);
}